// GAT_48679159333674
// MI455X (gfx1250) — compile-verified
//
#include <hip/hip_runtime.h>
#include <math.h>

#define NNODE 50000

typedef float v2f __attribute__((ext_vector_type(2)));
typedef float v8f __attribute__((ext_vector_type(8)));

// ---------------------------------------------------------------- utilities

__global__ void fill_kernel(float* __restrict__ p, float v, int n) {
    int i = blockIdx.x * blockDim.x + threadIdx.x;
    if (i < n) p[i] = v;
}

__device__ __forceinline__ void edge_nodes(const int* __restrict__ ei, int E,
                                           int e, int& s, int& d) {
    if (e < E) { s = ei[e]; d = ei[E + e]; }
    else       { s = e - E; d = s; }        // self loops appended after real edges
}

// float atomic max via signed-max / unsigned-min trick (cell init to -inf)
__device__ __forceinline__ void atomicMaxF(float* addr, float val) {
    if (val >= 0.0f) atomicMax((int*)addr, __float_as_int(val));
    else             atomicMin((unsigned int*)addr, __float_as_uint(val));
}

// ---------------------------------------------------------------- WMMA GEMM
// H[n, M] = X[n, K] @ W[K, M]; n % 16 == 0, M % 16 == 0, K % 4 == 0.
// One wave computes one 16x16 tile with V_WMMA_F32_16X16X4_F32 over K/4 steps.
__global__ void gemm_wmma_f32(const float* __restrict__ X,
                              const float* __restrict__ W,
                              float* __restrict__ H,
                              int n, int K, int M) {
    const int lane = threadIdx.x & 31;
    const int wave = threadIdx.x >> 5;
    const int wavesPerBlock = blockDim.x >> 5;

    const int colTiles = M >> 4;
    const int rowTiles = n >> 4;
    const int tile = blockIdx.x * wavesPerBlock + wave;
    if (tile >= rowTiles * colTiles) return;   // whole-wave uniform exit

    const int rowT = tile / colTiles;
    const int colT = tile - rowT * colTiles;
    const int row0 = rowT << 4;
    const int col0 = colT << 4;

    const int m16  = lane & 15;   // A: row M, B/D: col N
    const int half = lane >> 4;   // lane group 0/1

    v8f acc = {};
    const float* xrow = X + (size_t)(row0 + m16) * K;

    for (int k0 = 0; k0 < K; k0 += 4) {
        const int k = k0 + half * 2;          // ISA 16x4 A layout: k = 2*half + vgpr
        v2f a, b;
        a.x = xrow[k];
        a.y = xrow[k + 1];
        b.x = W[(size_t)(k)     * M + col0 + m16];   // B 4x16: mirrored layout
        b.y = W[(size_t)(k + 1) * M + col0 + m16];
        acc = __builtin_amdgcn_wmma_f32_16x16x4_f32(
                  /*neg_a=*/false, a, /*neg_b=*/false, b,
                  /*c_mod=*/(short)0, acc, /*reuse_a=*/false, /*reuse_b=*/false);
    }

    // C/D layout: VGPR v -> row (v + 8*half), col = lane & 15
#pragma unroll
    for (int v = 0; v < 8; ++v)
        H[(size_t)(row0 + half * 8 + v) * M + col0 + m16] = acc[v];
}

// ------------------------------------------------------- attention logits
// als[n,h] = sum_c H[n,h,c]*a_src[h,c]; ald likewise.
__global__ void att_logits(const float* __restrict__ H,
                           const float* __restrict__ a_src,
                           const float* __restrict__ a_dst,
                           float* __restrict__ als, float* __restrict__ ald,
                           int n, int heads, int C) {
    int t = blockIdx.x * blockDim.x + threadIdx.x;
    if (t >= n * heads) return;
    int node = t / heads;
    int h    = t - node * heads;
    const float* hp = H + (size_t)node * heads * C + (size_t)h * C;
    const float* as = a_src + h * C;
    const float* ad = a_dst + h * C;
    float s = 0.0f, d = 0.0f;
    for (int c = 0; c < C; ++c) { float v = hp[c]; s += v * as[c]; d += v * ad[c]; }
    als[t] = s;
    ald[t] = d;
}

// ----------------------------------------------------- segment max (pass 1)
__global__ void edge_max(const int* __restrict__ ei, int E, int Etot,
                         const float* __restrict__ als, const float* __restrict__ ald,
                         float* __restrict__ mbuf, int heads) {
    int e = blockIdx.x * blockDim.x + threadIdx.x;
    if (e >= Etot) return;
    int s, d;
    edge_nodes(ei, E, e, s, d);
    for (int h = 0; h < heads; ++h) {
        float x  = als[s * heads + h] + ald[d * heads + h];
        float lr = x >= 0.0f ? x : 0.2f * x;      // leaky_relu(0.2)
        atomicMaxF(&mbuf[d * heads + h], lr);
    }
}

// ------------------------------------- exp + denominator + numerator (pass 2)
__global__ void edge_acc(const int* __restrict__ ei, int E, int Etot,
                         const float* __restrict__ als, const float* __restrict__ ald,
                         const float* __restrict__ mbuf,
                         const float* __restrict__ H,
                         float* __restrict__ den, float* __restrict__ accum,
                         int heads, int C) {
    int t = blockIdx.x * blockDim.x + threadIdx.x;
    if (t >= Etot * heads) return;
    int e = t / heads;
    int h = t - e * heads;
    int s, d;
    edge_nodes(ei, E, e, s, d);
    float x  = als[s * heads + h] + ald[d * heads + h];
    float lr = x >= 0.0f ? x : 0.2f * x;
    float ex = __expf(lr - mbuf[d * heads + h]);
    atomicAdd(&den[d * heads + h], ex);
    const float* hs = H     + (size_t)s * heads * C + (size_t)h * C;
    float*       ap = accum + (size_t)d * heads * C + (size_t)h * C;
    for (int c = 0; c < C; ++c) atomicAdd(&ap[c], hs[c] * ex);
}

// --------------------------------------------------- alpha-normalize + bias
__global__ void finalize(float* __restrict__ accum, const float* __restrict__ den,
                         const float* __restrict__ bias,
                         int n, int heads, int C, int relu) {
    int HC = heads * C;
    int t  = blockIdx.x * blockDim.x + threadIdx.x;
    if (t >= n * HC) return;
    int node = t / HC;
    int rem  = t - node * HC;
    int h    = rem / C;
    float v = accum[t] / den[node * heads + h] + bias[rem];
    if (relu) v = fmaxf(v, 0.0f);
    accum[t] = v;
}

// ------------------------------------------------------------------ driver

extern "C" void kernel_launch(void* const* d_in, const int* in_sizes, int n_in,
                              void* d_out, int out_size, void* d_ws, size_t ws_size,
                              hipStream_t stream) {
    const float* x      = (const float*)d_in[0];
    const int*   ei     = (const int*)  d_in[1];   // [2,E] flat: [0..E)=src, [E..2E)=dst
    const float* W1     = (const float*)d_in[2];
    const float* a_src1 = (const float*)d_in[3];
    const float* a_dst1 = (const float*)d_in[4];
    const float* b1     = (const float*)d_in[5];
    const float* W2     = (const float*)d_in[6];
    const float* a_src2 = (const float*)d_in[7];
    const float* a_dst2 = (const float*)d_in[8];
    const float* b2     = (const float*)d_in[9];
    const float* W3     = (const float*)d_in[10];
    const float* a_src3 = (const float*)d_in[11];
    const float* a_dst3 = (const float*)d_in[12];
    const float* b3     = (const float*)d_in[13];

    const int E    = in_sizes[1] / 2;
    const int Etot = E + NNODE;

    // workspace layout (floats)
    float* ws   = (float*)d_ws;
    float* bufH = ws;                                  // N*240 (h1 then h3)
    float* act1 = bufH + (size_t)NNODE * 240;          // N*112
    float* h2   = act1 + (size_t)NNODE * 112;          // N*96
    float* act2 = h2   + (size_t)NNODE * 96;           // N*96
    float* als  = act2 + (size_t)NNODE * 96;           // N*7 (max heads)
    float* ald  = als  + (size_t)NNODE * 7;            // N*7
    float* mbuf = ald  + (size_t)NNODE * 7;            // N*7
    float* den  = mbuf + (size_t)NNODE * 7;            // N*7
    float* out  = (float*)d_out;                       // N*240 (layer-3 accumulator)

    auto layer = [&](const float* X, int K,
                     const float* W, const float* as, const float* ad, const float* b,
                     int heads, int C, float* Hbuf, float* accbuf, int relu) {
        const int HC = heads * C;

        // 1. H = X @ W  (WMMA, one wave per 16x16 tile, 8 waves / 256-thread block)
        const int tiles = (NNODE / 16) * (HC / 16);
        gemm_wmma_f32<<<(tiles + 7) / 8, 256, 0, stream>>>(X, W, Hbuf, NNODE, K, HC);

        // 2. per-node logits
        const int nh = NNODE * heads;
        att_logits<<<(nh + 255) / 256, 256, 0, stream>>>(Hbuf, as, ad, als, ald,
                                                         NNODE, heads, C);

        // 3. init segment-softmax state
        fill_kernel<<<(nh + 255) / 256, 256, 0, stream>>>(mbuf, -INFINITY, nh);
        hipMemsetAsync(den, 0, (size_t)nh * sizeof(float), stream);
        hipMemsetAsync(accbuf, 0, (size_t)NNODE * HC * sizeof(float), stream);

        // 4. segment max over incoming edges
        edge_max<<<(Etot + 255) / 256, 256, 0, stream>>>(ei, E, Etot, als, ald,
                                                         mbuf, heads);

        // 5. exp / denominator / weighted numerator scatter
        const int th = Etot * heads;
        edge_acc<<<(th + 255) / 256, 256, 0, stream>>>(ei, E, Etot, als, ald, mbuf,
                                                       Hbuf, den, accbuf, heads, C);

        // 6. normalize + bias (+ReLU), in place
        const int tot = NNODE * HC;
        finalize<<<(tot + 255) / 256, 256, 0, stream>>>(accbuf, den, b,
                                                        NNODE, heads, C, relu);
    };

    // Layer 1: [N,128] -> 7 heads x 16, ReLU
    layer(x,    128, W1, a_src1, a_dst1, b1, 7, 16, bufH, act1, 1);
    // Layer 2: [N,112] -> 6 heads x 16, ReLU
    layer(act1, 112, W2, a_src2, a_dst2, b2, 6, 16, h2,   act2, 1);
    // Layer 3: [N,96] -> 6 heads x 40, accumulate straight into d_out
    layer(act2,  96, W3, a_src3, a_dst3, b3, 6, 40, bufH, out,  0);
}